// LFQLayer_16552803959242
// MI455X (gfx1250) — compile-verified
//
#include <hip/hip_runtime.h>
#include <math.h>

typedef __attribute__((ext_vector_type(2))) float v2f;
typedef __attribute__((ext_vector_type(8))) float v8f;
typedef __attribute__((ext_vector_type(4))) int   v4i;

#define ROWS 4096
#define DIM  512
#define DD   14
#define RPB  128
#define NBLK (ROWS / RPB)
#define CODEK 16384

#if __has_builtin(__builtin_amdgcn_global_load_async_to_lds_b128) && \
    __has_builtin(__builtin_amdgcn_s_wait_asynccnt)
#define USE_ASYNC_LDS 1
#define AS_GLOBAL __attribute__((address_space(1)))
#define AS_LDS    __attribute__((address_space(3)))
#endif

__device__ __forceinline__ float hbent(float p) {
    float q = 1.0f - p;
    return -(p * logf(fmaxf(p, 1e-20f)) + q * logf(fmaxf(q, 1e-20f)));
}

__global__ __launch_bounds__(256) void lfq_init(float* ws) {
    int i = blockIdx.x * 256 + threadIdx.x;
    if (i < CODEK + 2) ws[i] = 0.0f;
}

__global__ __launch_bounds__(256) void lfq_main(
    const float* __restrict__ x, const float* __restrict__ w_in,
    const float* __restrict__ b_in, const float* __restrict__ w_out,
    const float* __restrict__ b_out, float* __restrict__ out,
    float* __restrict__ ws)
{
    __shared__ float s_ovl[16 * 512];  // 32KB overlay: w_in pad / prob halves / w_out pad
    __shared__ float s_z[RPB * 16];    // 8KB: z tile, cols 14,15 = 0
    __shared__ float s_bin[16];
    __shared__ float s_bout[512];

    const int tid  = threadIdx.x;
    const int wv   = tid >> 5;
    const int lane = tid & 31;
    const int hl   = lane >> 4;   // half-wave select
    const int ml   = lane & 15;
    const int rb0  = blockIdx.x * RPB;

    // stage w_in padded to [512][16]
    for (int i = tid; i < 512 * 16; i += 256) {
        int k = i >> 4, n = i & 15;
        s_ovl[i] = (n < DD) ? w_in[k * DD + n] : 0.0f;
    }
    if (tid < 16) s_bin[tid] = (tid < DD) ? b_in[tid] : 0.0f;
    for (int i = tid; i < 512; i += 256) s_bout[i] = b_out[i];
    __syncthreads();

    // ---- phase 1: z = x @ w_in  (M=16 rows/wave, N=16, K=512 via 16x16x4 f32 WMMA)
    {
        const int rbase = rb0 + wv * 16;
        const float* xr = x + (size_t)(rbase + ml) * DIM;
        v8f zacc = {0, 0, 0, 0, 0, 0, 0, 0};
#pragma unroll 4
        for (int k = 0; k < DIM; k += 4) {
            v2f a = *((const v2f*)(xr + k + 2 * hl));   // A: M=ml, K=k+2*hl+{0,1}
            v2f b;
            b[0] = s_ovl[(k + 2 * hl) * 16 + ml];       // B: N=ml, K matching A
            b[1] = s_ovl[(k + 2 * hl + 1) * 16 + ml];
            zacc = __builtin_amdgcn_wmma_f32_16x16x4_f32(false, a, false, b,
                                                         (short)0, zacc, false, false);
        }
        for (int r = 0; r < 8; r++)
            s_z[(wv * 16 + r + 8 * hl) * 16 + ml] = zacc[r] + s_bin[ml];
    }
    __syncthreads();

    // ---- phase 1b: indices, per-sample entropy (factorized), commit loss
    if (tid < RPB) {
        int row = tid, idx = 0;
        float H = 0.0f, com = 0.0f;
        for (int d = 0; d < DD; d++) {
            float zv = s_z[row * 16 + d];
            int bit = zv > 0.0f;
            idx |= bit << (13 - d);
            float s = bit ? 1.0f : -1.0f;
            float dq = zv - s;
            com += dq * dq;
            float p = 1.0f / (1.0f + expf(-400.0f * zv));
            H += hbent(p);
        }
        out[(size_t)ROWS * DIM + rb0 + row] = (float)idx;
        atomicAdd(&ws[CODEK + 0], H);
        atomicAdd(&ws[CODEK + 1], com);
    }

    // ---- phase 2: avg_prob accumulation Q[128x128] += A^T B over this block's rows
    v8f qacc[8];
    const v8f vz = {0, 0, 0, 0, 0, 0, 0, 0};
    for (int t = 0; t < 8; t++) qacc[t] = vz;

    for (int chunk = 0; chunk < 8; chunk++) {
        __syncthreads();   // prior overlay consumers done
        {
            int rowh = tid >> 3;
            int row  = rowh & 15;
            int half = rowh >> 4;            // 0: high 7 bits, 1: low 7 bits
            int c0   = (tid & 7) << 4;
            float p[7];
            int zb = (chunk * 16 + row) * 16 + half * 7;
            for (int dd = 0; dd < 7; dd++)
                p[dd] = 1.0f / (1.0f + expf(-400.0f * s_z[zb + dd]));
            float* dst = s_ovl + half * 2048 + row * 128;
            for (int c = c0; c < c0 + 16; c++) {
                float v = 1.0f;
                for (int dd = 0; dd < 7; dd++)
                    v *= ((c >> (6 - dd)) & 1) ? p[dd] : (1.0f - p[dd]);
                dst[c] = v;
            }
        }
        __syncthreads();
        // GEMM: M = codeA strip (wv*16), N = all 128 codeB, K = 16 chunk rows
        for (int ks = 0; ks < 4; ks++) {
            int ra = 4 * ks + 2 * hl;
            v2f a;
            a[0] = s_ovl[ra * 128 + wv * 16 + ml];
            a[1] = s_ovl[(ra + 1) * 128 + wv * 16 + ml];
            for (int nt = 0; nt < 8; nt++) {
                v2f b;
                b[0] = s_ovl[2048 + ra * 128 + nt * 16 + ml];
                b[1] = s_ovl[2048 + (ra + 1) * 128 + nt * 16 + ml];
                qacc[nt] = __builtin_amdgcn_wmma_f32_16x16x4_f32(false, a, false, b,
                                                                 (short)0, qacc[nt], false, false);
            }
        }
    }
    for (int nt = 0; nt < 8; nt++)
        for (int r = 0; r < 8; r++) {
            int m = wv * 16 + r + 8 * hl;
            int n = nt * 16 + ml;
            atomicAdd(&ws[m * 128 + n], qacc[nt][r]);
        }

    // ---- phase 3: out = quant @ w_out + b_out
    __syncthreads();
#ifdef USE_ASYNC_LDS
    {
        // padded [16][512]: rows 0..13 are exactly w_out's contiguous 7168 floats.
        // Copy via per-lane async b128 (ASYNCcnt path), zero the 2 pad rows normally.
        for (int i = tid; i < 1792; i += 256) {
            __builtin_amdgcn_global_load_async_to_lds_b128(
                (AS_GLOBAL v4i*)(w_out + i * 4),
                (AS_LDS v4i*)(s_ovl + i * 4),
                0, 0);
        }
        for (int i = tid; i < 1024; i += 256) s_ovl[7168 + i] = 0.0f;
        __builtin_amdgcn_s_wait_asynccnt(0);
    }
#else
    for (int i = tid; i < 16 * 512; i += 256) {
        int k = i >> 9, n = i & 511;
        s_ovl[i] = (k < DD) ? w_out[k * DIM + n] : 0.0f;
    }
#endif
    __syncthreads();
    {
        v2f aq[4];
        for (int ks = 0; ks < 4; ks++) {
            int kc = 4 * ks + 2 * hl;
            float z0 = s_z[(wv * 16 + ml) * 16 + kc];
            float z1 = s_z[(wv * 16 + ml) * 16 + kc + 1];
            aq[ks][0] = (kc     < DD) ? (z0 > 0.0f ? 1.0f : -1.0f) : 0.0f;
            aq[ks][1] = (kc + 1 < DD) ? (z1 > 0.0f ? 1.0f : -1.0f) : 0.0f;
        }
        const int rbase = rb0 + wv * 16;
        for (int nt = 0; nt < 32; nt++) {
            v8f o = vz;
            for (int ks = 0; ks < 4; ks++) {
                int kc = 4 * ks + 2 * hl;
                v2f b;
                b[0] = s_ovl[kc * 512 + nt * 16 + ml];
                b[1] = s_ovl[(kc + 1) * 512 + nt * 16 + ml];
                o = __builtin_amdgcn_wmma_f32_16x16x4_f32(false, aq[ks], false, b,
                                                          (short)0, o, false, false);
            }
            float bo = s_bout[nt * 16 + ml];
            for (int r = 0; r < 8; r++)
                out[(size_t)(rbase + r + 8 * hl) * DIM + nt * 16 + ml] = o[r] + bo;
        }
    }
}

__global__ __launch_bounds__(256) void lfq_finalize(const float* __restrict__ ws,
                                                    float* __restrict__ out) {
    __shared__ float red[256];
    float acc = 0.0f;
    for (int i = threadIdx.x; i < CODEK; i += 256) {
        float q = ws[i] * (1.0f / 4096.0f);
        acc += -q * logf(fmaxf(q, 1e-20f));
    }
    red[threadIdx.x] = acc;
    __syncthreads();
    for (int s = 128; s > 0; s >>= 1) {
        if (threadIdx.x < s) red[threadIdx.x] += red[threadIdx.x + s];
        __syncthreads();
    }
    if (threadIdx.x == 0) {
        float cbH  = red[0];
        float psum = ws[CODEK + 0] * (1.0f / 4096.0f);
        float com  = ws[CODEK + 1] * (1.0f / (4096.0f * 14.0f));
        float aux  = (psum - cbH) * 0.1f + com * 0.25f;
        out[(size_t)ROWS * DIM + ROWS] = aux;
    }
}

extern "C" void kernel_launch(void* const* d_in, const int* in_sizes, int n_in,
                              void* d_out, int out_size, void* d_ws, size_t ws_size,
                              hipStream_t stream) {
    const float* x     = (const float*)d_in[0];
    const float* w_in  = (const float*)d_in[1];
    const float* b_in  = (const float*)d_in[2];
    const float* w_out = (const float*)d_in[3];
    const float* b_out = (const float*)d_in[4];
    float* out = (float*)d_out;
    float* ws  = (float*)d_ws;

    lfq_init<<<(CODEK + 2 + 255) / 256, 256, 0, stream>>>(ws);
    lfq_main<<<NBLK, 256, 0, stream>>>(x, w_in, b_in, w_out, b_out, out, ws);
    lfq_finalize<<<1, 256, 0, stream>>>(ws, out);
}